// SpectralConv2d_4080218931943
// MI455X (gfx1250) — compile-verified
//
#include <hip/hip_runtime.h>
#include <hip/hip_bf16.h>
#include <cstdint>

// ---------------------------------------------------------------------------
// SpectralConv2d on MI455X (gfx1250, wave32, WMMA, async-to-LDS).
//
// H=W=421 is prime and only (20x20 + 20x20) spectral modes are used, so
// rfft2/irfft2 are replaced by partial DFTs expressed as fp32 WMMA GEMMs:
//   S1: Xw = x * Cw^T        (107776 x 421) x (421 x 40)   WMMA f32 16x16x4
//   S2: Xf = H-DFT(Xw)       small, L2-resident, VALU
//   S3: Of = Xf x W(gabor)   800 complex 8x32x32 matmuls   WMMA f32 16x16x4
//   S4: Yh = inv-H-DFT(Of)   small, L2-resident, VALU
//   S5: out = Yh * T5^T      (107776 x 40) x (40 x 421)    WMMA f32 16x16x4
// T5 folds 1/(H*W), the 2*cos/-2*sin irfft folding, and drops Im(k=0).
// Twiddle matrices are stored K-contiguous (transposed) so each WMMA B
// fragment pair is one aligned ds_load_b64; LDS strides padded for banks.
// ---------------------------------------------------------------------------

typedef float v2f __attribute__((ext_vector_type(2)));
typedef float v8f __attribute__((ext_vector_type(8)));

#define HH 421
#define WW 421
#define BB 8
#define CIN 32
#define COUT 32
#define NMODE 800           // 40 ky-modes * 20 kx-modes
#define NROW (BB*CIN*HH)    // 107776 = 6736 * 16
#define KP 448              // stage-1 K padded (8 chunks of 56)
#define NPAD 48             // 40 spectral cols (20 Re + 20 Im) padded
#define CHUNK 56
#define S1STR 60            // LDS inner stride (bank-conflict-free, 16B mult)
#define S3STR 34
#define S5STR 42

// ------------------- CDNA5 async global->LDS (guarded) ---------------------
#if __has_builtin(__builtin_amdgcn_global_load_async_to_lds_b32) && \
    __has_builtin(__builtin_amdgcn_global_load_async_to_lds_b128) && \
    __has_builtin(__builtin_amdgcn_s_wait_asynccnt)
#define HAS_ASYNC 1
#else
#define HAS_ASYNC 0
#endif

// Builtin signatures (from the compiler diagnostic):
//   b32 : (__as(1) int*,  __as(3) int*,  imm int, imm int)
//   b128: (__as(1) v4i*,  __as(3) v4i*,  imm int, imm int)
typedef int v4i __attribute__((vector_size(16)));
typedef __attribute__((address_space(1))) int  g_i32;
typedef __attribute__((address_space(3))) int  l_i32;
typedef __attribute__((address_space(1))) v4i  g_v4i;
typedef __attribute__((address_space(3))) v4i  l_v4i;

__device__ __forceinline__ void copy_b32_g2l(const void* g, void* l) {
#if HAS_ASYNC
  // flat LDS address truncates to the wave-relative LDS byte offset (ISA 10.2)
  __builtin_amdgcn_global_load_async_to_lds_b32(
      (g_i32*)(uintptr_t)g,
      (l_i32*)(uintptr_t)(unsigned int)(uintptr_t)l, 0, 0);
#else
  *(float*)l = *(const float*)g;
#endif
}
__device__ __forceinline__ void copy_b128_g2l(const void* g, void* l) {
#if HAS_ASYNC
  __builtin_amdgcn_global_load_async_to_lds_b128(
      (g_v4i*)(uintptr_t)g,
      (l_v4i*)(uintptr_t)(unsigned int)(uintptr_t)l, 0, 0);
#else
  *(float4*)l = *(const float4*)g;
#endif
}
__device__ __forceinline__ void async_wait0() {
#if HAS_ASYNC
  __builtin_amdgcn_s_wait_asynccnt(0);
#endif
}

__device__ __forceinline__ v8f wmma4(v2f a, v2f b, v8f c) {
  // V_WMMA_F32_16X16X4_F32 : D = A(16x4) x B(4x16) + C, fp32 throughout
  return __builtin_amdgcn_wmma_f32_16x16x4_f32(
      false, a, false, b, (short)0, c, false, false);
}

// --------------------------- precompute kernels ----------------------------

// Cwt[j(48)][k(448)] : transposed W-DFT matrix, K-contiguous
__global__ void k_fill_cw(float* __restrict__ Cwt) {
  int t = blockIdx.x * 256 + threadIdx.x;
  if (t >= NPAD * KP) return;
  int j = t / KP, r = t % KP;
  float v = 0.f;
  if (r < WW && j < 40) {
    const float TWO_PI = 6.28318530717958647692f;
    int kx = (j < 20) ? j : (j - 20);
    int ph = (kx * r) % WW;                 // exact phase reduction
    float th = TWO_PI * (float)ph / (float)WW;
    v = (j < 20) ? cosf(th) : -sinf(th);    // e^{-i th}: (cos, -sin)
  }
  Cwt[t] = v;
}

__global__ void k_fill_ht(float* __restrict__ cosT, float* __restrict__ sinT) {
  int t = blockIdx.x * 256 + threadIdx.x;
  if (t >= HH * 40) return;
  int h = t / 40, a = t % 40;
  int ky = (a < 20) ? a : (HH - 20 + (a - 20));   // modes {0..19, 401..420}
  int ph = (int)(((long)ky * h) % HH);
  const float TWO_PI = 6.28318530717958647692f;
  float th = TWO_PI * (float)ph / (float)HH;
  cosT[t] = cosf(th);
  sinT[t] = sinf(th);
}

// T5t[w(432)][j(42)] : transposed irfft fold matrix, K-contiguous
__global__ void k_fill_t5(float* __restrict__ T5t) {
  int t = blockIdx.x * 256 + threadIdx.x;
  if (t >= 432 * S5STR) return;
  int w = t / S5STR, j = t % S5STR;
  float v = 0.f;
  const float inv = 1.0f / ((float)HH * (float)WW);
  const float TWO_PI = 6.28318530717958647692f;
  if (w < WW && j < 40) {
    if (j == 0) {
      v = inv;                                  // DC: Re only
    } else if (j < 20) {
      int ph = (j * w) % WW;
      v = 2.f * inv * cosf(TWO_PI * (float)ph / (float)WW);
    } else if (j == 20) {
      v = 0.f;                                  // irfft discards Im(k=0)
    } else {
      int ph = ((j - 20) * w) % WW;
      v = -2.f * inv * sinf(TWO_PI * (float)ph / (float)WW);
    }
  }
  T5t[t] = v;
}

// WRe/WIm layout: [mode][o][i]  (K=i contiguous -> b64 B-fragment loads)
__global__ void k_fill_w(float* __restrict__ WRe, float* __restrict__ WIm,
                         const float* __restrict__ w1re, const float* __restrict__ w1im,
                         const float* __restrict__ w2re, const float* __restrict__ w2im,
                         const float* __restrict__ pfreq, const float* __restrict__ ptheta,
                         const float* __restrict__ psigma, const float* __restrict__ pgamma) {
  int t = blockIdx.x * 256 + threadIdx.x;
  if (t >= NMODE * 1024) return;
  int mode = t >> 10;
  int io = t & 1023;
  int o = io >> 5, i = io & 31;                // i fastest -> contiguous stores
  int a = mode / 20, kx = mode % 20;
  float freq = pfreq[0], th = ptheta[0], sg = psigma[0], gm = pgamma[0];
  // ifftshift of 430-grid: gs[r][c] = g[(r+215)%430][(c+215)%430]
  // g1 rows 0..19 -> grid rows 215..234 ; g2 rows (410..429) -> 195..214
  int gr = (a < 20) ? (a + 215) : (a + 175);
  int gc = kx + 215;
  const float step = 1.0f / 429.0f;            // linspace(-0.5,0.5,430)
  float y  = -0.5f + (float)gr * step;
  float xg = -0.5f + (float)gc * step;
  float ct = cosf(th), st = sinf(th);
  float u = xg * ct + y * st;
  float v = -xg * st + y * ct;
  float sx = sg, sy = sg / (gm + 1e-5f);
  const float PI = 3.14159265358979323846f;
  float du = u - freq / PI;
  float t1 = sx * sx * du * du;
  float t2 = sy * sy * v * v;
  float g = expf(-2.0f * PI * PI * (t1 + t2));
  int r = (a < 20) ? a : (a - 20);
  long widx = ((long)(i * COUT + o) * 20 + r) * 20 + kx;
  float wr = (a < 20 ? w1re : w2re)[widx];
  float wi = (a < 20 ? w1im : w2im)[widx];
  WRe[t] = wr * g;
  WIm[t] = wi * g;
}

// --------------------- stage 1: W-direction DFT (WMMA) ---------------------
// Xw[row][48]: cols 0..19 = Re(kx), 20..39 = Im(kx), 40..47 = 0
__global__ void k_dft_w(const float* __restrict__ x, const float* __restrict__ Cwt,
                        float* __restrict__ Xw) {
  __shared__ __align__(16) float ldsA[8][16][S1STR];   // per-wave x tile chunk
  __shared__ __align__(16) float ldsBt[NPAD][S1STR];   // twiddle chunk [n][k]
  const int wave = threadIdx.x >> 5, lane = threadIdx.x & 31;
  const int m = lane & 15, khalf = lane >> 4;
  const long row0 = ((long)blockIdx.x * 8 + wave) * 16;
  v8f acc0 = {}, acc1 = {}, acc2 = {};
  for (int kc = 0; kc < KP; kc += CHUNK) {
    __syncthreads();
    // B chunk: ldsBt[n][c] = Cwt[n][kc+c]; contiguous 16B-aligned segments
    for (int idx = threadIdx.x; idx < NPAD * (CHUNK / 4); idx += 256) {
      int n = idx / (CHUNK / 4), q = idx % (CHUNK / 4);
      copy_b128_g2l(Cwt + (long)n * KP + kc + 4 * q, &ldsBt[n][4 * q]);
    }
    // A chunk (x rows are only 4B aligned -> b32)
    for (int idx = lane; idx < 16 * CHUNK; idx += 32) {
      int r = idx / CHUNK, c = idx % CHUNK, gk = kc + c;
      const float* xrow = x + (row0 + r) * WW;
      if (gk < WW) {
        copy_b32_g2l(xrow + gk, &ldsA[wave][r][c]);
        if (gk + CHUNK < WW) __builtin_prefetch(&xrow[gk + CHUNK], 0, 0);
      } else {
        ldsA[wave][r][c] = 0.f;
      }
    }
    async_wait0();
    __syncthreads();
#pragma unroll
    for (int s = 0; s < CHUNK / 4; ++s) {
      int k = 4 * s + 2 * khalf;
      v2f a;  a.x  = ldsA[wave][m][k];   a.y  = ldsA[wave][m][k + 1];
      v2f b0; b0.x = ldsBt[m][k];        b0.y = ldsBt[m][k + 1];
      v2f b1; b1.x = ldsBt[16 + m][k];   b1.y = ldsBt[16 + m][k + 1];
      v2f b2; b2.x = ldsBt[32 + m][k];   b2.y = ldsBt[32 + m][k + 1];
      acc0 = wmma4(a, b0, acc0);
      acc1 = wmma4(a, b1, acc1);
      acc2 = wmma4(a, b2, acc2);
    }
  }
#pragma unroll
  for (int v = 0; v < 8; ++v) {
    long r = row0 + v + 8 * khalf;
    Xw[r * NPAD + m]       = acc0[v];
    Xw[r * NPAD + 16 + m]  = acc1[v];
    Xw[r * NPAD + 32 + m]  = acc2[v];
  }
}

// ------------------- stage 2: H-direction DFT (L2-bound) -------------------
__global__ void k_dft_h(const float* __restrict__ Xw,
                        const float* __restrict__ cosT, const float* __restrict__ sinT,
                        float* __restrict__ XfRe, float* __restrict__ XfIm) {
  int t = blockIdx.x * 256 + threadIdx.x;        // 256*40*20 = 204800 exact
  int kx = t % 20;
  int a = (t / 20) % 40;
  int bi = t / 800;
  const float* base = Xw + (long)bi * HH * NPAD;
  float re = 0.f, im = 0.f;
  for (int h = 0; h < HH; ++h) {
    float c = cosT[h * 40 + a], s = sinT[h * 40 + a];
    float xr = base[h * NPAD + kx], xi = base[h * NPAD + 20 + kx];
    re += xr * c + xi * s;                       // x * e^{-i th}
    im += xi * c - xr * s;
  }
  XfRe[bi * NMODE + a * 20 + kx] = re;
  XfIm[bi * NMODE + a * 20 + kx] = im;
}

// --------------- stage 3: per-mode complex CI->CO mix (WMMA) ---------------
__global__ void k_mix(const float* __restrict__ XfRe, const float* __restrict__ XfIm,
                      const float* __restrict__ WRe, const float* __restrict__ WIm,
                      float* __restrict__ OfRe, float* __restrict__ OfIm) {
  __shared__ __align__(16) float lAr[4][16][S3STR];   // rows 8..15 zero
  __shared__ __align__(16) float lAi[4][16][S3STR];
  const int t = threadIdx.x;
  for (int idx = t; idx < 4096; idx += 256) {
    int mm = idx >> 10, rem = idx & 1023;
    int isIm = rem >> 9, rem2 = rem & 511;
    int r = rem2 >> 5, i = rem2 & 31;
    int mode = blockIdx.x * 4 + mm;
    float v = 0.f;
    if (r < BB) {
      const float* src = isIm ? XfIm : XfRe;
      v = src[(r * CIN + i) * NMODE + mode];
    }
    (isIm ? lAi : lAr)[mm][r][i] = v;
  }
  __syncthreads();
  const int wave = t >> 5, lane = t & 31;
  const int wm = wave >> 1;                       // mode-in-block (2 waves/mode)
  const int n0 = (wave & 1) * 16;                 // output-channel tile
  const int m = lane & 15, khalf = lane >> 4;
  const int mode = blockIdx.x * 4 + wm;
  const float* brb = WRe + (long)mode * 1024;     // [o][i], K contiguous
  const float* bib = WIm + (long)mode * 1024;
  v8f aRR = {}, aII = {}, aRI = {}, aIR = {};
#pragma unroll
  for (int s = 0; s < 8; ++s) {                   // K = CI = 32
    int k = 4 * s + 2 * khalf;
    v2f ar; ar.x = lAr[wm][m][k];  ar.y = lAr[wm][m][k + 1];
    v2f ai; ai.x = lAi[wm][m][k];  ai.y = lAi[wm][m][k + 1];
    v2f br; br.x = brb[(n0 + m) * 32 + k];  br.y = brb[(n0 + m) * 32 + k + 1];
    v2f bi; bi.x = bib[(n0 + m) * 32 + k];  bi.y = bib[(n0 + m) * 32 + k + 1];
    aRR = wmma4(ar, br, aRR);
    aII = wmma4(ai, bi, aII);
    aRI = wmma4(ar, bi, aRI);
    aIR = wmma4(ai, br, aIR);
  }
  if (khalf == 0) {                               // rows 0..7 = batch
#pragma unroll
    for (int v = 0; v < 8; ++v) {
      int b = v, o = n0 + m;
      long idx = (long)(b * COUT + o) * NMODE + mode;
      OfRe[idx] = aRR[v] - aII[v];                // complex matmul combine
      OfIm[idx] = aRI[v] + aIR[v];
    }
  }
}

// --------------- stage 4: inverse H-direction DFT (L2-bound) ---------------
__global__ void k_idft_h(const float* __restrict__ OfRe, const float* __restrict__ OfIm,
                         const float* __restrict__ cosT, const float* __restrict__ sinT,
                         float* __restrict__ Yh) {
  long t = (long)blockIdx.x * 256 + threadIdx.x;  // 256*421*20 = 2155520 exact
  if (t >= (long)BB * COUT * HH * 20) return;
  int kx = (int)(t % 20);
  int h = (int)((t / 20) % HH);
  int bo = (int)(t / (HH * 20));
  const float* pre = OfRe + (long)bo * NMODE + kx;
  const float* pim = OfIm + (long)bo * NMODE + kx;
  float yr = 0.f, yi = 0.f;
  for (int a = 0; a < 40; ++a) {
    float c = cosT[h * 40 + a], s = sinT[h * 40 + a];
    float orr = pre[a * 20], oii = pim[a * 20];
    yr += orr * c - oii * s;                      // O * e^{+i th}
    yi += orr * s + oii * c;
  }
  long row = (long)bo * HH + h;
  Yh[row * NPAD + kx] = yr;
  Yh[row * NPAD + 20 + kx] = yi;
}

// --------------- stage 5: inverse W-direction (real out, WMMA) -------------
__global__ void k_idft_w(const float* __restrict__ Yh, const float* __restrict__ T5t,
                         float* __restrict__ out) {
  __shared__ __align__(16) float ldsBt[432][S5STR];    // 72.6 KB, K-contiguous
  // async bulk copy of the whole fold matrix (b128, 16B-aligned both sides)
  for (int idx = threadIdx.x; idx < (432 * S5STR) / 4; idx += 256)
    copy_b128_g2l(T5t + 4 * idx, ((float*)ldsBt) + 4 * idx);
  async_wait0();
  __syncthreads();
  const int wave = threadIdx.x >> 5, lane = threadIdx.x & 31;
  const int m = lane & 15, khalf = lane >> 4;
  const long row0 = ((long)blockIdx.x * 8 + wave) * 16;
  // A fragments (K=40) kept in registers, reused across all 27 N-tiles.
  v2f a[10];
  const float* arow = Yh + (row0 + m) * NPAD;     // Yh is L2-hot (~21 MB)
#pragma unroll
  for (int s = 0; s < 10; ++s) {
    int k = 4 * s + 2 * khalf;
    a[s].x = arow[k]; a[s].y = arow[k + 1];
  }
  for (int nt = 0; nt < 27; ++nt) {
    v8f acc = {};
    int n0 = nt * 16;
#pragma unroll
    for (int s = 0; s < 10; ++s) {
      int k = 4 * s + 2 * khalf;
      v2f b; b.x = ldsBt[n0 + m][k]; b.y = ldsBt[n0 + m][k + 1];
      acc = wmma4(a[s], b, acc);
    }
    int w = n0 + m;
    if (w < WW) {
#pragma unroll
      for (int v = 0; v < 8; ++v) {
        long r = row0 + v + 8 * khalf;
        out[r * WW + w] = acc[v];
      }
    }
  }
}

// ------------------------------- launcher ----------------------------------

extern "C" void kernel_launch(void* const* d_in, const int* in_sizes, int n_in,
                              void* d_out, int out_size, void* d_ws, size_t ws_size,
                              hipStream_t stream) {
  const float* x    = (const float*)d_in[0];
  const float* w1re = (const float*)d_in[1];
  const float* w1im = (const float*)d_in[2];
  const float* w2re = (const float*)d_in[3];
  const float* w2im = (const float*)d_in[4];
  const float* freq = (const float*)d_in[5];
  const float* thet = (const float*)d_in[6];
  const float* sigm = (const float*)d_in[7];
  const float* gamm = (const float*)d_in[8];
  float* out = (float*)d_out;

  float* ws = (float*)d_ws;
  size_t off = 0;
  auto alloc = [&](size_t n) {
    float* p = ws + off;
    off += (n + 63) & ~(size_t)63;
    return p;
  };
  float* Cwt  = alloc((size_t)NPAD * KP);        //  21.5 K f
  float* cosT = alloc((size_t)HH * 40);          //  16.8 K f
  float* sinT = alloc((size_t)HH * 40);
  float* T5t  = alloc((size_t)432 * S5STR);
  float* WRe  = alloc((size_t)NMODE * 1024);     // 819.2 K f
  float* WIm  = alloc((size_t)NMODE * 1024);
  float* XwYh = alloc((size_t)NROW * NPAD);      // 5.17 M f (S1->S2, then S4->S5)
  float* XfRe = alloc((size_t)BB * CIN * NMODE); // 204.8 K f
  float* XfIm = alloc((size_t)BB * CIN * NMODE);
  float* OfRe = alloc((size_t)BB * COUT * NMODE);
  float* OfIm = alloc((size_t)BB * COUT * NMODE);
  (void)in_sizes; (void)n_in; (void)out_size; (void)ws_size;

  // precompute twiddles / gabor-modulated weights
  k_fill_cw<<<(NPAD * KP + 255) / 256, 256, 0, stream>>>(Cwt);
  k_fill_ht<<<(HH * 40 + 255) / 256, 256, 0, stream>>>(cosT, sinT);
  k_fill_t5<<<(432 * S5STR + 255) / 256, 256, 0, stream>>>(T5t);
  k_fill_w<<<(NMODE * 1024) / 256, 256, 0, stream>>>(WRe, WIm, w1re, w1im, w2re, w2im,
                                                     freq, thet, sigm, gamm);
  // pipeline
  k_dft_w <<<NROW / 16 / 8, 256, 0, stream>>>(x, Cwt, XwYh);
  k_dft_h <<<(BB * CIN * 40 * 20) / 256, 256, 0, stream>>>(XwYh, cosT, sinT, XfRe, XfIm);
  k_mix   <<<NMODE / 4, 256, 0, stream>>>(XfRe, XfIm, WRe, WIm, OfRe, OfIm);
  k_idft_h<<<((long)BB * COUT * HH * 20 + 255) / 256, 256, 0, stream>>>(OfRe, OfIm, cosT,
                                                                        sinT, XwYh);
  k_idft_w<<<NROW / 16 / 8, 256, 0, stream>>>(XwYh, T5t, out);
}